// Part_wise_EMD_51522427683007
// MI455X (gfx1250) — compile-verified
//
#include <hip/hip_runtime.h>

typedef float v2f __attribute__((ext_vector_type(2)));
typedef float v8f __attribute__((ext_vector_type(8)));

#define TOTAL_V 991
#define NPARTS  20
#define NP      96      // padded problem size (max n = 92)
#define CST     97      // LDS row stride (odd -> conflict-free column walks)
#define NBATCH  2048
#define BIGC    1.0e30f
#define NEGH   -3.0e38f

__constant__ int c_off[NPARTS + 1] = {
    0, 45, 106, 149, 194, 286, 320, 361, 423, 467, 511,
    569, 611, 651, 711, 752, 787, 851, 879, 929, 991};

__constant__ float c_eps[9] = {
    64.0f, 16.0f, 4.0f, 1.0f, 0.25f, 0.0625f,
    0.015625f, 0.00390625f, 0.0025f};

__device__ __forceinline__ float wave_max(float v) {
#pragma unroll
  for (int o = 16; o > 0; o >>= 1) v = fmaxf(v, __shfl_xor(v, o, 32));
  return v;
}
__device__ __forceinline__ float wave_sum(float v) {
#pragma unroll
  for (int o = 16; o > 0; o >>= 1) v += __shfl_xor(v, o, 32);
  return v;
}

__global__ __launch_bounds__(128)
void sinkhorn_part_kernel(const float* __restrict__ X,   // predvert
                          const float* __restrict__ Y,   // batch_pointclouds
                          float* __restrict__ partial) {
  const int part  = blockIdx.x;
  const int batch = blockIdx.y;
  const int s = c_off[part];
  const int n = c_off[part + 1] - s;

  __shared__ float xs[NP * 4], ys[NP * 4];     // padded xyz0 per point
  __shared__ float hx[NP], hy[NP];             // 0.5 * squared norms
  __shared__ float C[NP * CST];                // Cxy cost matrix
  __shared__ float f[NP], g[NP], px[NP], py[NP];
  __shared__ float ft[NP], gt[NP], pxt[NP], pyt[NP];
  __shared__ float red[4];

  const int tid  = threadIdx.x;
  const int lane = tid & 31;
  const int wv   = tid >> 5;

  // ---- Phase 0: stage points into LDS (zero-padded to 96x4) ----
  if (tid < NP) {
    float a0 = 0.f, a1 = 0.f, a2 = 0.f, b0 = 0.f, b1 = 0.f, b2 = 0.f;
    if (tid < n) {
      size_t base = ((size_t)batch * TOTAL_V + s + tid) * 3;
      a0 = X[base]; a1 = X[base + 1]; a2 = X[base + 2];
      b0 = Y[base]; b1 = Y[base + 1]; b2 = Y[base + 2];
    }
    xs[tid * 4 + 0] = a0; xs[tid * 4 + 1] = a1;
    xs[tid * 4 + 2] = a2; xs[tid * 4 + 3] = 0.f;
    ys[tid * 4 + 0] = b0; ys[tid * 4 + 1] = b1;
    ys[tid * 4 + 2] = b2; ys[tid * 4 + 3] = 0.f;
    hx[tid] = 0.5f * (a0 * a0 + a1 * a1 + a2 * a2);
    hy[tid] = 0.5f * (b0 * b0 + b1 * b1 + b2 * b2);
    f[tid] = 0.f; g[tid] = 0.f; px[tid] = 0.f; py[tid] = 0.f;
  }
  __syncthreads();

  // ---- Phase 1: Cxy = hx[i] + hy[j] - X.Y^T via WMMA f32 16x16x4 ----
  // 6x6 tiles of 16x16, K=4 (3 coords + zero pad); 9 tiles per wave.
  for (int t = wv; t < 36; t += 4) {
    const int ti = t / 6, tj = t % 6;
    const int r  = lane & 15;
    const int kb = (lane < 16) ? 0 : 2;   // K pair held by this half-wave
    v2f a, b;
    a.x = xs[(ti * 16 + r) * 4 + kb];
    a.y = xs[(ti * 16 + r) * 4 + kb + 1];
    b.x = ys[(tj * 16 + r) * 4 + kb];
    b.y = ys[(tj * 16 + r) * 4 + kb + 1];
    v8f acc = {};
    acc = __builtin_amdgcn_wmma_f32_16x16x4_f32(
        false, a, false, b, (short)0, acc, false, false);

    const int  N     = tj * 16 + (lane & 15);
    const int  Mbase = ti * 16 + ((lane < 16) ? 0 : 8);
    const float hyN  = hy[N];
    const bool ncol  = (N < n);
    // Branch-free masked epilogue: loads hoisted, mask -> v_cndmask.
    float hxq[8];
#pragma unroll
    for (int q = 0; q < 8; ++q) hxq[q] = hx[Mbase + q];   // contiguous -> b128
#pragma unroll
    for (int q = 0; q < 8; ++q) {
      const int M = Mbase + q;
      float val   = hxq[q] + hyN - acc[q];
      val = (ncol && (M < n)) ? val : BIGC;
      C[M * CST + N] = val;
    }
  }
  __syncthreads();

  // ---- Phase 2: 9 eps-scaling rounds, 4 LSE sweeps each ----
  const float lab = -__logf((float)n);   // loga == logb (n == m)
#pragma unroll 1
  for (int e = 0; e < 9; ++e) {
    const float eps = c_eps[e];
    const float ie  = 1.0f / eps;
    // 4 reduction families x 96 rows = 384 wave-tasks, round-robin over 4 waves
#pragma unroll 1
    for (int tk = wv; tk < 4 * NP; tk += 4) {
      const int which = tk / NP;
      const int idx   = tk - which * NP;
      const int j0 = lane, j1 = lane + 32, j2 = lane + 64;
      float t0, t1, t2;
      if (which == 0) {            // ft[i] : row LSE of (g - Cxy)/eps + logb
        const float* row = &C[idx * CST];
        t0 = (g[j0] - row[j0]) * ie + lab;
        t1 = (g[j1] - row[j1]) * ie + lab;
        t2 = (g[j2] - row[j2]) * ie + lab;
      } else if (which == 1) {     // gt[j] : col LSE of (f - Cxy)/eps + loga
        t0 = (f[j0] - C[j0 * CST + idx]) * ie + lab;
        t1 = (f[j1] - C[j1 * CST + idx]) * ie + lab;
        t2 = (f[j2] - C[j2 * CST + idx]) * ie + lab;
      } else if (which == 2) {     // px row: Cxx on the fly
        const float xi0 = xs[idx * 4], xi1 = xs[idx * 4 + 1],
                    xi2 = xs[idx * 4 + 2], hxi = hx[idx];
        float d0 = xi0 * xs[j0 * 4] + xi1 * xs[j0 * 4 + 1] + xi2 * xs[j0 * 4 + 2];
        float d1 = xi0 * xs[j1 * 4] + xi1 * xs[j1 * 4 + 1] + xi2 * xs[j1 * 4 + 2];
        float d2 = xi0 * xs[j2 * 4] + xi1 * xs[j2 * 4 + 1] + xi2 * xs[j2 * 4 + 2];
        t0 = (j0 < n) ? (px[j0] - (hxi + hx[j0] - d0)) * ie + lab : NEGH;
        t1 = (j1 < n) ? (px[j1] - (hxi + hx[j1] - d1)) * ie + lab : NEGH;
        t2 = (j2 < n) ? (px[j2] - (hxi + hx[j2] - d2)) * ie + lab : NEGH;
      } else {                     // py row: Cyy on the fly
        const float yi0 = ys[idx * 4], yi1 = ys[idx * 4 + 1],
                    yi2 = ys[idx * 4 + 2], hyi = hy[idx];
        float d0 = yi0 * ys[j0 * 4] + yi1 * ys[j0 * 4 + 1] + yi2 * ys[j0 * 4 + 2];
        float d1 = yi0 * ys[j1 * 4] + yi1 * ys[j1 * 4 + 1] + yi2 * ys[j1 * 4 + 2];
        float d2 = yi0 * ys[j2 * 4] + yi1 * ys[j2 * 4 + 1] + yi2 * ys[j2 * 4 + 2];
        t0 = (j0 < n) ? (py[j0] - (hyi + hy[j0] - d0)) * ie + lab : NEGH;
        t1 = (j1 < n) ? (py[j1] - (hyi + hy[j1] - d1)) * ie + lab : NEGH;
        t2 = (j2 < n) ? (py[j2] - (hyi + hy[j2] - d2)) * ie + lab : NEGH;
      }
      float m  = wave_max(fmaxf(t0, fmaxf(t1, t2)));
      float sm = wave_sum(__expf(t0 - m) + __expf(t1 - m) + __expf(t2 - m));
      float res = (idx < n) ? -eps * (m + __logf(sm)) : 0.f;  // pin padding to 0
      if (lane == 0) {
        if      (which == 0) ft[idx]  = res;
        else if (which == 1) gt[idx]  = res;
        else if (which == 2) pxt[idx] = res;
        else                 pyt[idx] = res;
      }
    }
    __syncthreads();
    if (tid < NP) {
      f[tid]  = 0.5f * (f[tid]  + ft[tid]);
      g[tid]  = 0.5f * (g[tid]  + gt[tid]);
      px[tid] = 0.5f * (px[tid] + pxt[tid]);
      py[tid] = 0.5f * (py[tid] + pyt[tid]);
    }
    __syncthreads();
  }

  // ---- Phase 3: loss = mean(f - px) + mean(g - py) ----
  float v = 0.f;
  if (tid < n) v = (f[tid] - px[tid]) + (g[tid] - py[tid]);
  v = wave_sum(v);
  if (lane == 0) red[wv] = v;
  __syncthreads();
  if (tid == 0) {
    float tot = red[0] + red[1] + red[2] + red[3];
    partial[(size_t)batch * NPARTS + part] = tot / ((float)n * (float)NBATCH);
  }
}

__global__ __launch_bounds__(256)
void final_reduce_kernel(const float* __restrict__ p, float* __restrict__ out,
                         int count) {
  __shared__ float red[8];
  float v = 0.f;
  for (int i = threadIdx.x; i < count; i += 256) v += p[i];
  v = wave_sum(v);
  const int lane = threadIdx.x & 31, wv = threadIdx.x >> 5;
  if (lane == 0) red[wv] = v;
  __syncthreads();
  if (threadIdx.x == 0) {
    float t = 0.f;
#pragma unroll
    for (int i = 0; i < 8; ++i) t += red[i];
    out[0] = t;
  }
}

extern "C" void kernel_launch(void* const* d_in, const int* in_sizes, int n_in,
                              void* d_out, int out_size, void* d_ws,
                              size_t ws_size, hipStream_t stream) {
  const float* X = (const float*)d_in[0];   // predvert
  const float* Y = (const float*)d_in[1];   // batch_pointclouds
  float* out = (float*)d_out;
  float* ws  = (float*)d_ws;                // 40960 partial sums

  dim3 grid(NPARTS, NBATCH);
  sinkhorn_part_kernel<<<grid, 128, 0, stream>>>(X, Y, ws);
  final_reduce_kernel<<<1, 256, 0, stream>>>(ws, out, NPARTS * NBATCH);
}